// GeneralLayer_61847529063045
// MI455X (gfx1250) — compile-verified
//
#include <hip/hip_runtime.h>
#include <math.h>

typedef __attribute__((ext_vector_type(2))) float v2f;
typedef __attribute__((ext_vector_type(8))) float v8f;

#define FDIM 128
#define BN_EPS_F 1e-5f
#define L2_EPS_F 1e-12f
#define MT_PER_WG 4   // M-tiles (16 rows each) per workgroup in the GEMM

// ---------------------------------------------------------------------------
// 1) zero scratch (grid-stride, float4)
__global__ __launch_bounds__(256) void zero_kernel(float* p, size_t n4) {
    size_t i = (size_t)blockIdx.x * blockDim.x + threadIdx.x;
    size_t stride = (size_t)gridDim.x * blockDim.x;
    float4 z = make_float4(0.f, 0.f, 0.f, 0.f);
    for (; i < n4; i += stride) ((float4*)p)[i] = z;
}

// ---------------------------------------------------------------------------
// 2) degree accumulation: one thread per edge
__global__ __launch_bounds__(256) void degree_kernel(const int* __restrict__ src,
                                                     const int* __restrict__ dst,
                                                     float* deg_out, float* deg_in, int E) {
    int e = blockIdx.x * blockDim.x + threadIdx.x;
    if (e >= E) return;
    atomicAdd(&deg_out[src[e]], 1.0f);
    atomicAdd(&deg_in[dst[e]], 1.0f);
}

// ---------------------------------------------------------------------------
// 3) rsqrt(max(deg,1))
__global__ __launch_bounds__(256) void rinv_kernel(const float* __restrict__ deg_out,
                                                   const float* __restrict__ deg_in,
                                                   float* rinv_out, float* rinv_in, int N) {
    int i = blockIdx.x * blockDim.x + threadIdx.x;
    if (i >= N) return;
    rinv_out[i] = rsqrtf(fmaxf(deg_out[i], 1.0f));
    rinv_in[i]  = rsqrtf(fmaxf(deg_in[i], 1.0f));
}

// ---------------------------------------------------------------------------
// 4) edge scatter: one wave per edge, lane covers 4 features (128 = 32*4)
__global__ __launch_bounds__(256) void scatter_kernel(const float* __restrict__ h,
                                                      const int* __restrict__ src,
                                                      const int* __restrict__ dst,
                                                      const float* __restrict__ rinv_out,
                                                      float* agg, int E) {
    int e = blockIdx.x * 8 + (threadIdx.x >> 5);
    if (e >= E) return;
    int lane = threadIdx.x & 31;
    int s = src[e];
    int d = dst[e];
    float sc = rinv_out[s];
    const float4 hv = *(const float4*)(h + (size_t)s * FDIM + lane * 4);
    float* out = agg + (size_t)d * FDIM + lane * 4;
    atomicAdd(out + 0, hv.x * sc);
    atomicAdd(out + 1, hv.y * sc);
    atomicAdd(out + 2, hv.z * sc);
    atomicAdd(out + 3, hv.w * sc);
}

// ---------------------------------------------------------------------------
// 5) WMMA GEMM: out_raw = (agg * rsqrt(deg_in)) @ W, fused BN column stats.
//    Workgroup = 8 waves (256 threads). Wave w owns N-tile n0 = w*16 and
//    iterates over MT_PER_WG M-tiles, reusing its preloaded B fragments.
//    V_WMMA_F32_16X16X4_F32 layouts (ISA 7.12.2):
//      A (16x4):  lanes 0-15 -> K=0(v0),1(v1); lanes 16-31 -> K=2(v0),3(v1); M=lane%16
//      B (4x16):  v0/v1 hold K rows 2*khalf / 2*khalf+1; N=lane%16
//      C/D:       VGPR r -> row r (lanes 0-15) / row r+8 (lanes 16-31); N=lane%16
__global__ __launch_bounds__(256) void gemm_wmma_kernel(const float* __restrict__ agg,
                                                        const float* __restrict__ rinv_in,
                                                        const float* __restrict__ Wm,
                                                        float* __restrict__ out_raw,
                                                        float* colsum, float* colsumsq,
                                                        int N) {
    const int wave  = threadIdx.x >> 5;   // 0..7
    const int lane  = threadIdx.x & 31;
    const int ln    = lane & 15;          // N / M index within tile
    const int khalf = lane >> 4;          // 0 or 1: which K-pair this lane holds
    const int n0    = wave * 16;
    const int m_base = blockIdx.x * (16 * MT_PER_WG);

    // Preload B fragments for all 32 K-steps (K = 4*s .. 4*s+3)
    v2f b[32];
#pragma unroll
    for (int s = 0; s < 32; ++s) {
        int k = 4 * s + 2 * khalf;
        b[s].x = Wm[(size_t)k * FDIM + n0 + ln];
        b[s].y = Wm[(size_t)(k + 1) * FDIM + n0 + ln];
    }

    for (int t = 0; t < MT_PER_WG; ++t) {
        int m0 = m_base + t * 16;
        if (m0 >= N) break;               // wave-uniform: EXEC stays all-ones
        int row = m0 + ln;
        float rscale = rinv_in[row];
        const float* arow = agg + (size_t)row * FDIM + 2 * khalf;

        v8f c = {};
#pragma unroll
        for (int s = 0; s < 32; ++s) {
            v2f a = *(const v2f*)(arow + 4 * s);
            a = a * rscale;
            c = __builtin_amdgcn_wmma_f32_16x16x4_f32(
                    false, a, false, b[s], (short)0, c, false, false);
        }

        // Store D and accumulate BN partial sums for column n0+ln
        float s1 = 0.f, s2 = 0.f;
#pragma unroll
        for (int r = 0; r < 8; ++r) {
            float v = c[r];
            out_raw[(size_t)(m0 + khalf * 8 + r) * FDIM + n0 + ln] = v;
            s1 += v;
            s2 += v * v;
        }
        s1 += __shfl_xor(s1, 16, 32);     // combine rows 0-7 with rows 8-15
        s2 += __shfl_xor(s2, 16, 32);
        if (khalf == 0) {
            atomicAdd(&colsum[n0 + ln], s1);
            atomicAdd(&colsumsq[n0 + ln], s2);
        }
    }
}

// ---------------------------------------------------------------------------
// 6) BN finalize: fold gamma/beta/mu/sigma into per-column affine a,b
__global__ __launch_bounds__(128) void bn_finalize_kernel(const float* __restrict__ colsum,
                                                          const float* __restrict__ colsumsq,
                                                          const float* __restrict__ gamma,
                                                          const float* __restrict__ beta,
                                                          float* bn_a, float* bn_b, int N) {
    int i = threadIdx.x;
    float invN = 1.0f / (float)N;
    float mu = colsum[i] * invN;
    float var = colsumsq[i] * invN - mu * mu;
    float is = rsqrtf(var + BN_EPS_F);
    float a = gamma[i] * is;
    bn_a[i] = a;
    bn_b[i] = beta[i] - mu * a;
}

// ---------------------------------------------------------------------------
// 7) per-row: affine + ReLU + L2 normalize (one wave per row, in place)
__global__ __launch_bounds__(256) void finish_kernel(float* __restrict__ out,
                                                     const float* __restrict__ bn_a,
                                                     const float* __restrict__ bn_b,
                                                     int N) {
    int row = blockIdx.x * 8 + (threadIdx.x >> 5);
    if (row >= N) return;
    int lane = threadIdx.x & 31;
    float* p = out + (size_t)row * FDIM + lane * 4;
    float4 v = *(const float4*)p;
    const float4 a = *(const float4*)(bn_a + lane * 4);
    const float4 b = *(const float4*)(bn_b + lane * 4);
    v.x = fmaxf(v.x * a.x + b.x, 0.f);
    v.y = fmaxf(v.y * a.y + b.y, 0.f);
    v.z = fmaxf(v.z * a.z + b.z, 0.f);
    v.w = fmaxf(v.w * a.w + b.w, 0.f);
    float ss = v.x * v.x + v.y * v.y + v.z * v.z + v.w * v.w;
#pragma unroll
    for (int m = 16; m >= 1; m >>= 1) ss += __shfl_xor(ss, m, 32);
    float scale = 1.0f / fmaxf(sqrtf(ss), L2_EPS_F);
    v.x *= scale; v.y *= scale; v.z *= scale; v.w *= scale;
    *(float4*)p = v;
}

// ---------------------------------------------------------------------------
extern "C" void kernel_launch(void* const* d_in, const int* in_sizes, int n_in,
                              void* d_out, int out_size, void* d_ws, size_t ws_size,
                              hipStream_t stream) {
    const float* h     = (const float*)d_in[0];
    const float* Wm    = (const float*)d_in[1];
    const float* gamma = (const float*)d_in[2];
    const float* beta  = (const float*)d_in[3];
    const int*   src   = (const int*)d_in[4];
    const int*   dst   = (const int*)d_in[5];
    const int N = in_sizes[0] / FDIM;   // 50000
    const int E = in_sizes[4];          // 600000

    // workspace layout (floats)
    float* ws       = (float*)d_ws;
    float* deg_out  = ws;
    float* deg_in   = deg_out + N;
    float* rinv_out = deg_in + N;
    float* rinv_in  = rinv_out + N;
    float* colsum   = rinv_in + N;
    float* colsumsq = colsum + FDIM;
    float* bn_a     = colsumsq + FDIM;
    float* bn_b     = bn_a + FDIM;
    float* agg      = bn_b + FDIM;
    float* out_raw  = (float*)d_out;    // raw GEMM output lives in d_out

    // 1) zero [deg_out .. agg end]  (deg/rinv/colsums/bn + agg)
    size_t zero_floats = (size_t)4 * N + 4 * FDIM + (size_t)N * FDIM;
    size_t zero4 = (zero_floats + 3) / 4;
    int zgrid = (int)((zero4 + 255) / 256);
    if (zgrid > 4096) zgrid = 4096;
    zero_kernel<<<zgrid, 256, 0, stream>>>(ws, zero4);

    // 2) degrees
    degree_kernel<<<(E + 255) / 256, 256, 0, stream>>>(src, dst, deg_out, deg_in, E);

    // 3) rsqrt(max(deg,1))
    rinv_kernel<<<(N + 255) / 256, 256, 0, stream>>>(deg_out, deg_in, rinv_out, rinv_in, N);

    // 4) edge scatter (one wave per edge)
    scatter_kernel<<<(E + 7) / 8, 256, 0, stream>>>(h, src, dst, rinv_out, agg, E);

    // 5) WMMA GEMM + BN stats
    int mtiles = (N + 15) / 16;
    int wgs = (mtiles + MT_PER_WG - 1) / MT_PER_WG;
    gemm_wmma_kernel<<<wgs, 256, 0, stream>>>(agg, rinv_in, Wm, out_raw, colsum, colsumsq, N);

    // 6) BN finalize
    bn_finalize_kernel<<<1, 128, 0, stream>>>(colsum, colsumsq, gamma, beta, bn_a, bn_b, N);

    // 7) affine + ReLU + L2 normalize, in place on d_out
    finish_kernel<<<(N + 7) / 8, 256, 0, stream>>>(out_raw, bn_a, bn_b, N);
}